// SupConLoss_53738630808173
// MI455X (gfx1250) — compile-verified
//
#include <hip/hip_runtime.h>
#include <hip/hip_bf16.h>
#include <math.h>

typedef __attribute__((ext_vector_type(2))) float v2f;
typedef __attribute__((ext_vector_type(8))) float v8f;

// Types for the async global->LDS builtin (param is AS1/AS3 pointer to int4).
typedef int v4i_t __attribute__((vector_size(16)));
typedef __attribute__((address_space(1))) v4i_t* gas_v4p;
typedef __attribute__((address_space(3))) v4i_t* las_v4p;

#define WAVE 32
#define DIMC 128              // feature dim (reference: D = 128)
#define KSTEPS (DIMC / 4)     // 32 wmma f32 16x16x4 steps per tile
#define LDS_STRIDE 132        // 128 + 4 pad -> lane m starts at bank 4m (conflict-free)

#if defined(__has_builtin)
#if __has_builtin(__builtin_amdgcn_global_load_async_to_lds_b128) && \
    __has_builtin(__builtin_amdgcn_s_wait_asynccnt)
#define USE_ASYNC_LDS 1
#endif
#endif

// ---------------------------------------------------------------------------
// Kernel 1: L2-normalize rows. One wave per row of [N, 128].
// ---------------------------------------------------------------------------
__global__ __launch_bounds__(256) void supcon_normalize(
    const float* __restrict__ emb, float* __restrict__ Fn, int N) {
  int wave = (blockIdx.x * blockDim.x + threadIdx.x) / WAVE;
  int lane = threadIdx.x & (WAVE - 1);
  if (wave >= N) return;
  const float4* src = (const float4*)(emb + (size_t)wave * DIMC);
  float4 v = src[lane];                       // 32 lanes * 4 floats = 128
  float ss = v.x * v.x + v.y * v.y + v.z * v.z + v.w * v.w;
  #pragma unroll
  for (int off = 16; off >= 1; off >>= 1) ss += __shfl_xor(ss, off, WAVE);
  float nrm = sqrtf(ss);
  float s = 1.0f / fmaxf(nrm, 1e-12f);
  float4 o;
  o.x = v.x * s; o.y = v.y * s; o.z = v.z * s; o.w = v.w * s;
  ((float4*)(Fn + (size_t)wave * DIMC))[lane] = o;
}

// ---------------------------------------------------------------------------
// Kernel 2: fused Gram + row statistics, double-buffered j-panels.
// Grid: N/128 blocks of 256 threads (8 waves). Each wave owns a 16-row
// i-block (A panel held in registers); the workgroup streams all j-panels
// through LDS (padded stride, conflict-free ds_load_2addr_b64) while the
// next panel is fetched asynchronously (ASYNCcnt) or via register prefetch.
// ---------------------------------------------------------------------------
__global__ __launch_bounds__(256) void supcon_main(
    const float* __restrict__ Fn, const int* __restrict__ labels,
    const int* __restrict__ wmask, const float* __restrict__ temp,
    float* __restrict__ rowLoss, int N) {
  __shared__ float ldsB[2 * 16 * LDS_STRIDE];   // two 16x128 panels, padded

  const float invT = 1.0f / temp[0];
  const float S = invT;              // fixed shift: |logit| <= 1/temp

  const int tid  = threadIdx.x;
  const int wv   = tid >> 5;
  const int lane = tid & 31;
  const int half = lane >> 4;        // 0: K0/K1, C rows 0..7 ; 1: K2/K3, C rows 8..15
  const int m    = lane & 15;        // A: row in tile; B/C: column in tile
  const int i0   = (blockIdx.x * 8 + wv) * 16;

  // Panel-copy indices: 16x128 floats = 512 float4; thread handles q0,q1.
  const int q0 = tid, q1 = tid + 256;
  const int pr0 = q0 >> 5, pc0 = q0 & 31;       // row, float4-col
  const int pr1 = q1 >> 5, pc1 = q1 & 31;

  // Preload A panel (ISA 32-bit 16x4 A layout: lane<16 K=4k..4k+1, lane>=16 K=4k+2..4k+3)
  v2f a[KSTEPS];
  {
    const float* arow = Fn + (size_t)(i0 + m) * DIMC + 2 * half;
    #pragma unroll
    for (int kk = 0; kk < KSTEPS; ++kk) a[kk] = *(const v2f*)(arow + kk * 4);
  }

  // Row labels / valid for the 8 C rows this lane covers (row = 8*half + v)
  int   rlab[8];
  float rval[8];
  #pragma unroll
  for (int v = 0; v < 8; ++v) {
    int r = i0 + 8 * half + v;
    int lb = labels[r];
    rlab[v] = lb;
    rval[v] = (wmask[r] != 0 && lb != -1) ? 1.0f : 0.0f;
  }

  float sExp[8], sPos[8], sCnt[8];
  #pragma unroll
  for (int v = 0; v < 8; ++v) { sExp[v] = 0.0f; sPos[v] = 0.0f; sCnt[v] = 0.0f; }

  const int nTiles = N / 16;

#if defined(USE_ASYNC_LDS)
  // ---- async global->LDS panel pipeline (ASYNCcnt) ----
  auto issue_panel = [&](int p, int buf) {
    const float* g0 = Fn + (size_t)(p * 16 + pr0) * DIMC + pc0 * 4;
    const float* g1 = Fn + (size_t)(p * 16 + pr1) * DIMC + pc1 * 4;
    float* l0 = ldsB + buf * (16 * LDS_STRIDE) + pr0 * LDS_STRIDE + pc0 * 4;
    float* l1 = ldsB + buf * (16 * LDS_STRIDE) + pr1 * LDS_STRIDE + pc1 * 4;
    __builtin_amdgcn_global_load_async_to_lds_b128((gas_v4p)g0, (las_v4p)l0, 0, 0);
    __builtin_amdgcn_global_load_async_to_lds_b128((gas_v4p)g1, (las_v4p)l1, 0, 0);
  };
  issue_panel(0, 0);
#else
  // ---- register-prefetch panel pipeline (LOADcnt) ----
  float4 rg0, rg1;
  auto ld_regs = [&](int p) {
    rg0 = ((const float4*)(Fn + (size_t)(p * 16 + pr0) * DIMC))[pc0];
    rg1 = ((const float4*)(Fn + (size_t)(p * 16 + pr1) * DIMC))[pc1];
  };
  auto st_panel = [&](int buf) {
    *(float4*)(ldsB + buf * (16 * LDS_STRIDE) + pr0 * LDS_STRIDE + pc0 * 4) = rg0;
    *(float4*)(ldsB + buf * (16 * LDS_STRIDE) + pr1 * LDS_STRIDE + pc1 * 4) = rg1;
  };
  ld_regs(0);
  st_panel(0);
  ld_regs(1);            // prefetch panel 1 (nTiles >= 2 always here)
#endif

  for (int jt = 0; jt < nTiles; ++jt) {
    const int cur = jt & 1;

#if defined(USE_ASYNC_LDS)
    __builtin_amdgcn_s_wait_asynccnt(0);   // own panel DMAs landed
    __syncthreads();                       // everyone's landed; prev compute done
    if (jt + 1 < nTiles) issue_panel(jt + 1, cur ^ 1);  // overlap with compute
#else
    __syncthreads();                       // buf[cur] stores visible everywhere
    if (jt + 1 < nTiles) {
      st_panel(cur ^ 1);                   // stage prefetched panel
      if (jt + 2 < nTiles) ld_regs(jt + 2);
    }
#endif

    // 16x16 tile: two independent WMMA accumulation chains over K = 128.
    v8f c0 = {0.0f, 0.0f, 0.0f, 0.0f, 0.0f, 0.0f, 0.0f, 0.0f};
    v8f c1 = {0.0f, 0.0f, 0.0f, 0.0f, 0.0f, 0.0f, 0.0f, 0.0f};
    const float* brow = ldsB + cur * (16 * LDS_STRIDE) + m * LDS_STRIDE + 2 * half;
    #pragma unroll
    for (int kk = 0; kk < KSTEPS; kk += 2) {
      v2f b0 = *(const v2f*)(brow + kk * 4);
      v2f b1 = *(const v2f*)(brow + kk * 4 + 4);
      c0 = __builtin_amdgcn_wmma_f32_16x16x4_f32(
          false, a[kk], false, b0, (short)0, c0, false, false);
      c1 = __builtin_amdgcn_wmma_f32_16x16x4_f32(
          false, a[kk + 1], false, b1, (short)0, c1, false, false);
    }
    v8f c = c0 + c1;

    const int   j0   = jt * 16;
    const int   col  = j0 + m;
    const int   clab = labels[col];
    const float cval = (wmask[col] != 0 && clab != -1) ? 1.0f : 0.0f;

    #pragma unroll
    for (int v = 0; v < 8; ++v) {
      int   row   = i0 + 8 * half + v;
      float logit = c[v] * invT;
      float pv    = rval[v] * cval * ((row != col) ? 1.0f : 0.0f);
      sExp[v] += pv * __expf(logit - S);
      float isPos = (rlab[v] == clab) ? pv : 0.0f;
      sPos[v] += isPos * logit;
      sCnt[v] += isPos;
    }
  }

  // Reduce across the 16 lanes of each half (columns of the tile).
  #pragma unroll
  for (int v = 0; v < 8; ++v) {
    #pragma unroll
    for (int off = 8; off >= 1; off >>= 1) {
      sExp[v] += __shfl_xor(sExp[v], off, WAVE);
      sPos[v] += __shfl_xor(sPos[v], off, WAVE);
      sCnt[v] += __shfl_xor(sCnt[v], off, WAVE);
    }
  }
  if (m == 0) {  // lanes 0 and 16 hold totals for rows 0..7 / 8..15
    #pragma unroll
    for (int v = 0; v < 8; ++v) {
      int   row  = i0 + 8 * half + v;
      float lse  = S + __logf(sExp[v]);
      float mlpp = (sPos[v] - sCnt[v] * lse) / fmaxf(sCnt[v], 1e-6f);
      rowLoss[row] = mlpp * rval[v];
    }
  }
}

// ---------------------------------------------------------------------------
// Kernel 3: final reduction -> loss = -(sum rowLoss) / (sum valid)
// ---------------------------------------------------------------------------
__global__ __launch_bounds__(256) void supcon_finalize(
    const float* __restrict__ rowLoss, const int* __restrict__ labels,
    const int* __restrict__ wmask, float* __restrict__ out, int N) {
  __shared__ float sL[256];
  __shared__ float sV[256];
  float accL = 0.0f, accV = 0.0f;
  for (int i = threadIdx.x; i < N; i += 256) {
    accL += rowLoss[i];
    accV += (wmask[i] != 0 && labels[i] != -1) ? 1.0f : 0.0f;
  }
  sL[threadIdx.x] = accL;
  sV[threadIdx.x] = accV;
  __syncthreads();
  for (int s = 128; s > 0; s >>= 1) {
    if (threadIdx.x < s) {
      sL[threadIdx.x] += sL[threadIdx.x + s];
      sV[threadIdx.x] += sV[threadIdx.x + s];
    }
    __syncthreads();
  }
  if (threadIdx.x == 0) out[0] = -sL[0] / sV[0];
}

// ---------------------------------------------------------------------------
extern "C" void kernel_launch(void* const* d_in, const int* in_sizes, int n_in,
                              void* d_out, int out_size, void* d_ws, size_t ws_size,
                              hipStream_t stream) {
  const float* emb    = (const float*)d_in[0];
  const int*   labels = (const int*)d_in[1];
  const int*   wmask  = (const int*)d_in[2];
  const float* temp   = (const float*)d_in[3];

  const int N = in_sizes[1];       // 6144 anchors; D fixed at 128 (DIMC)

  float* Fn      = (float*)d_ws;                    // N*128 floats
  float* rowLoss = (float*)d_ws + (size_t)N * DIMC; // N floats

  supcon_normalize<<<N / 8, 256, 0, stream>>>(emb, Fn, N);
  supcon_main<<<N / 128, 256, 0, stream>>>(Fn, labels, wmask, temp, rowLoss, N);
  supcon_finalize<<<1, 256, 0, stream>>>(rowLoss, labels, wmask, (float*)d_out, N);
}